// GraphConv_5866925326658
// MI455X (gfx1250) — compile-verified
//
#include <hip/hip_runtime.h>

typedef __attribute__((ext_vector_type(2))) float v2f;
typedef __attribute__((ext_vector_type(8))) float v8f;

#define N_NODES 50000
#define N_EDGES 800000
#define D 64
#define N_TILES ((N_NODES / 16) * (D / 16)) // 3125 * 4 = 12500

// ---------------------------------------------------------------------------
// Kernel 1: zero the aggregation buffer (N_NODES x D floats) in d_ws.
// 3.2M floats = 800k float4 stores.
// ---------------------------------------------------------------------------
__global__ void gc_zero_agg(float4* __restrict__ agg4) {
    int i = blockIdx.x * blockDim.x + threadIdx.x;
    if (i < (N_NODES * D) / 4) {
        agg4[i] = make_float4(0.f, 0.f, 0.f, 0.f);
    }
}

// ---------------------------------------------------------------------------
// Kernel 2: edge scatter.  msgs = feat[src] - w  ;  agg[dst] += msgs
// One wave32 per edge; each lane handles a float2 (32 lanes x 2 = 64 feats).
// Gathers and atomics are L2-resident (feat/agg are 12.8 MB each << 192 MB L2).
// Non-returning atomicAdd -> global_atomic_add_f32 tracked on STOREcnt.
// ---------------------------------------------------------------------------
__global__ void gc_scatter(const float* __restrict__ feat,
                           const float* __restrict__ ew,
                           const int*   __restrict__ src,
                           const int*   __restrict__ dst,
                           float*       __restrict__ agg) {
    int gid  = blockIdx.x * blockDim.x + threadIdx.x;
    int edge = gid >> 5;
    int lane = threadIdx.x & 31;
    if (edge >= N_EDGES) return;

    int   s = src[edge];
    int   d = dst[edge];
    float w = ew[edge];

    const float2 f = ((const float2*)(feat + (long)s * D))[lane];
    float* ap = agg + (long)d * D + lane * 2;
    atomicAdd(ap + 0, f.x - w);
    atomicAdd(ap + 1, f.y - w);
}

// ---------------------------------------------------------------------------
// Kernel 3: out = feat @ W1 + agg @ W2 using V_WMMA_F32_16X16X4_F32.
// One wave per 16x16 output tile; K=64 -> 16 WMMA steps per matrix product.
//
// f32 A (16x4) layout:  lane = (half<<4)|m ; holds rows M=m, K = 2*half + {0,1}
// f32 B (4x16) layout:  mirrored: lane half selects K pair, m = column N
// f32 C/D (16x16):      VGPR v: lanes 0-15 -> (M=v, N=lane),
//                               lanes 16-31 -> (M=v+8, N=lane-16)
// ---------------------------------------------------------------------------
__global__ void gc_project(const float* __restrict__ feat,
                           const float* __restrict__ agg,
                           const float* __restrict__ W1,
                           const float* __restrict__ W2,
                           float*       __restrict__ out) {
    int wave = blockIdx.x * (blockDim.x >> 5) + (threadIdx.x >> 5);
    if (wave >= N_TILES) return;          // whole-wave guard: EXEC stays all-1s

    int rowTile = wave >> 2;              // D/16 = 4 column tiles
    int colTile = wave & 3;
    int row0 = rowTile << 4;
    int col0 = colTile << 4;

    int lane = threadIdx.x & 31;
    int m    = lane & 15;                 // A: row within tile / B: column
    int kk   = (lane >> 4) << 1;          // 0 for lanes 0-15, 2 for lanes 16-31

    const float* fa = feat + (long)(row0 + m) * D;
    const float* ga = agg  + (long)(row0 + m) * D;

    v8f c = {};

#pragma unroll
    for (int k0 = 0; k0 < D; k0 += 4) {
        v2f a, b;
        a.x = fa[k0 + kk];
        a.y = fa[k0 + kk + 1];
        b.x = W1[(k0 + kk)     * D + col0 + m];
        b.y = W1[(k0 + kk + 1) * D + col0 + m];
        c = __builtin_amdgcn_wmma_f32_16x16x4_f32(
                /*neg_a=*/false, a, /*neg_b=*/false, b,
                /*c_mod=*/(short)0, c, /*reuse_a=*/false, /*reuse_b=*/false);
    }

#pragma unroll
    for (int k0 = 0; k0 < D; k0 += 4) {
        v2f a, b;
        a.x = ga[k0 + kk];
        a.y = ga[k0 + kk + 1];
        b.x = W2[(k0 + kk)     * D + col0 + m];
        b.y = W2[(k0 + kk + 1) * D + col0 + m];
        c = __builtin_amdgcn_wmma_f32_16x16x4_f32(
                false, a, false, b, (short)0, c, false, false);
    }

    // Store D-matrix per the 32-bit 16x16 C/D layout.
    int mOff = (lane >> 4) << 3;          // lanes 16-31 hold M = v + 8
    float* op = out + col0 + m;
#pragma unroll
    for (int v = 0; v < 8; ++v) {
        op[(long)(row0 + v + mOff) * D] = c[v];
    }
}

// ---------------------------------------------------------------------------
// Host-side launcher.  Inputs (setup_inputs order):
//   0: feat [50000,64] f32   1: edge_weight [800000,1] f32
//   2: W1 [64,64] f32        3: W2 [64,64] f32
//   4: src [800000] int      5: dst [800000] int
// Output: [50000,64] f32.  Workspace: agg (12.8 MB) at start of d_ws.
// ---------------------------------------------------------------------------
extern "C" void kernel_launch(void* const* d_in, const int* in_sizes, int n_in,
                              void* d_out, int out_size, void* d_ws, size_t ws_size,
                              hipStream_t stream) {
    const float* feat = (const float*)d_in[0];
    const float* ew   = (const float*)d_in[1];
    const float* W1   = (const float*)d_in[2];
    const float* W2   = (const float*)d_in[3];
    const int*   src  = (const int*)d_in[4];
    const int*   dst  = (const int*)d_in[5];
    float* out = (float*)d_out;
    float* agg = (float*)d_ws;

    // 1) agg = 0
    {
        int n4 = (N_NODES * D) / 4;                     // 800000
        gc_zero_agg<<<(n4 + 255) / 256, 256, 0, stream>>>((float4*)agg);
    }
    // 2) scatter-add edges (wave per edge)
    {
        long threads = (long)N_EDGES * 32;              // 25.6M threads
        int blocks = (int)((threads + 255) / 256);      // 100000 blocks
        gc_scatter<<<blocks, 256, 0, stream>>>(feat, ew, src, dst, agg);
    }
    // 3) fused dual projection via f32 WMMA
    {
        int wavesPerBlock = 256 / 32;                   // 8
        int blocks = (N_TILES + wavesPerBlock - 1) / wavesPerBlock; // 1563
        gc_project<<<blocks, 256, 0, stream>>>(feat, agg, W1, W2, out);
    }
}